// SSNHead_75179107549593
// MI455X (gfx1250) — compile-verified
//
#include <hip/hip_runtime.h>
#include <hip/hip_bf16.h>

typedef __attribute__((ext_vector_type(2))) float v2f;
typedef __attribute__((ext_vector_type(8))) float v8f;

#define NPROP  8192
#define FEAT   1024
#define NSEG   9
#define TILE_M 16
// LDS row pitch in floats: 3072 + 4 pad so lane stride (mod 64 banks) != 0
#define PITCH  3076

__device__ __forceinline__ int iminc(int a, int b) { return a < b ? a : b; }

// Fused SSN head: STPP pooling + 3 linear heads, one pass over x.
__global__ __launch_bounds__(256)
void ssn_head_fused(const float* __restrict__ x,
                    const float* __restrict__ sf,
                    const float* __restrict__ w_act,  const float* __restrict__ b_act,
                    const float* __restrict__ w_comp, const float* __restrict__ b_comp,
                    const float* __restrict__ w_reg,  const float* __restrict__ b_reg,
                    float* __restrict__ out)
{
    extern __shared__ float q[];            // [TILE_M][PITCH] pooled features
    const int tid = threadIdx.x;
    const int n0  = blockIdx.x * TILE_M;    // first proposal of this tile

    // ---------------- Phase 1: stream x once, pool into LDS ----------------
    // q[n][0..1023]    = (x[s0]+x[s1]) * sf0/6     (starting:  /2 mean, /3 mult, *sf0)
    // q[n][1024..2047] = (x[s2..s6])   * 1/15      (course:    /5 mean, /3 mult)
    // q[n][2048..3071] = (x[s7]+x[s8]) * sf1/6     (ending)
    // activity_feat = course mean = 3 * q_part1 -> folded as 3x on the accumulator.
    for (int i = tid; i < TILE_M * (FEAT / 4); i += 256) {
        const int n  = i >> 8;              // 0..15
        const int d4 = (i & 255) << 2;      // 0,4,...,1020
        const float4* xb =
            (const float4*)(x + (size_t)(n0 + n) * NSEG * FEAT + d4);
        float4 s0 = xb[0 * 256], s1 = xb[1 * 256];
        float4 c2 = xb[2 * 256], c3 = xb[3 * 256], c4 = xb[4 * 256];
        float4 c5 = xb[5 * 256], c6 = xb[6 * 256];
        float4 e7 = xb[7 * 256], e8 = xb[8 * 256];

        const float sf0 = sf[(n0 + n) * 2 + 0] * (1.0f / 6.0f);
        const float sf1 = sf[(n0 + n) * 2 + 1] * (1.0f / 6.0f);
        const float cm  = 1.0f / 15.0f;

        float4 p0, p1, p2;
        p0.x = (s0.x + s1.x) * sf0;  p0.y = (s0.y + s1.y) * sf0;
        p0.z = (s0.z + s1.z) * sf0;  p0.w = (s0.w + s1.w) * sf0;
        p1.x = (c2.x + c3.x + c4.x + c5.x + c6.x) * cm;
        p1.y = (c2.y + c3.y + c4.y + c5.y + c6.y) * cm;
        p1.z = (c2.z + c3.z + c4.z + c5.z + c6.z) * cm;
        p1.w = (c2.w + c3.w + c4.w + c5.w + c6.w) * cm;
        p2.x = (e7.x + e8.x) * sf1;  p2.y = (e7.y + e8.y) * sf1;
        p2.z = (e7.z + e8.z) * sf1;  p2.w = (e7.w + e8.w) * sf1;

        float* qn = q + n * PITCH;
        *(float4*)(qn + d4)            = p0;
        *(float4*)(qn + FEAT + d4)     = p1;
        *(float4*)(qn + 2 * FEAT + d4) = p2;
    }

    __syncthreads();

    // ------------- Phase 2: WMMA f32 16x16x4 GEMM, one head-tile per wave -------------
    // wave 0-1: activity (21 ch, K = middle 1024 only, acc * 3)
    // wave 2-3: completeness (20 ch, K = 3072)
    // wave 4-6: regression   (40 ch, K = 3072)
    // wave 7  : idle in this phase
    // NOTE: wave id is forced into an SGPR via readfirstlane so all head selection is
    // scalar-uniform: constant trip counts, s_cbranch (EXEC stays all-ones for WMMA).
    const int wu   = __builtin_amdgcn_readfirstlane(tid >> 5);
    const int lane = tid & 31;
    const int m     = lane & 15;              // A row / B column within tile
    const int khalf = (lane >> 4) << 1;       // lanes 16..31 hold K=2,3
    const int rbase = n0 + ((lane >> 4) << 3);// hi lanes hold M=8..15

    if (wu < 2) {
        // ---- activity head: out = 3 * (q_mid @ w_act^T) + b_act ----
        const int cl = wu * 16 + m;
        const float* qs = q + m * PITCH + FEAT + khalf;
        const float* bp = w_act + (size_t)iminc(cl, 20) * FEAT + khalf;

        v8f acc = {};
        #pragma unroll 8
        for (int kk = 0; kk < FEAT; kk += 4) {
            v2f a = *(const v2f*)(qs + kk);      // ds_load_b64
            v2f b = *(const v2f*)(bp + kk);      // global_load_b64 (L2-hot)
            acc = __builtin_amdgcn_wmma_f32_16x16x4_f32(
                      false, a, false, b, (short)0, acc, false, false);
        }
        if (cl <= 20) {
            const float bias = b_act[cl];
            #pragma unroll
            for (int i = 0; i < 8; ++i)
                out[(size_t)(rbase + i) * 21 + cl] = acc[i] * 3.0f + bias;
        }
    } else if (wu < 7) {
        // ---- completeness / regression heads: full K = 3072 ----
        const bool iscomp = (wu < 4);
        const int  cl     = (iscomp ? (wu - 2) : (wu - 4)) * 16 + m;
        const int  cmax   = iscomp ? 19 : 39;
        const float* wb   = iscomp ? w_comp : w_reg;
        const float* qs   = q + m * PITCH + khalf;
        const float* bp   = wb + (size_t)iminc(cl, cmax) * 3 * FEAT + khalf;

        v8f acc = {};
        #pragma unroll 8
        for (int kk = 0; kk < 3 * FEAT; kk += 4) {
            v2f a = *(const v2f*)(qs + kk);      // ds_load_b64
            v2f b = *(const v2f*)(bp + kk);      // global_load_b64 (L2-hot)
            acc = __builtin_amdgcn_wmma_f32_16x16x4_f32(
                      false, a, false, b, (short)0, acc, false, false);
        }
        if (cl <= cmax) {
            const size_t obase  = iscomp ? (size_t)NPROP * 21 : (size_t)NPROP * 41;
            const int    opitch = iscomp ? 20 : 40;
            const float  bias   = iscomp ? b_comp[cl] : b_reg[cl];
            #pragma unroll
            for (int i = 0; i < 8; ++i)
                out[obase + (size_t)(rbase + i) * opitch + cl] = acc[i] + bias;
        }
    }
}

extern "C" void kernel_launch(void* const* d_in, const int* in_sizes, int n_in,
                              void* d_out, int out_size, void* d_ws, size_t ws_size,
                              hipStream_t stream) {
    const float* x      = (const float*)d_in[0];
    const float* sfac   = (const float*)d_in[1];
    const float* w_act  = (const float*)d_in[2];
    const float* b_act  = (const float*)d_in[3];
    const float* w_comp = (const float*)d_in[4];
    const float* b_comp = (const float*)d_in[5];
    const float* w_reg  = (const float*)d_in[6];
    const float* b_reg  = (const float*)d_in[7];
    float* out = (float*)d_out;

    const size_t smem = (size_t)TILE_M * PITCH * sizeof(float);  // ~197 KB (CDNA5: 320 KB/WGP)
    ssn_head_fused<<<dim3(NPROP / TILE_M), dim3(256), smem, stream>>>(
        x, sfac, w_act, b_act, w_comp, b_comp, w_reg, b_reg, out);
}